// SingleLayerRNNModel_61452392071465
// MI455X (gfx1250) — compile-verified
//
#include <hip/hip_runtime.h>
#include <stdint.h>

// Fused single-pass tanh-RNN for MI455X (gfx1250, wave32).
//   h_t = tanh([x_t | h_{t-1}] * [W_ih^T ; W_hh^T] + (b_ih+b_hh))
// K = IN+HID = 8 -> exactly two chained V_WMMA_F32_16X16X4_F32 per timestep.
// Each wave owns 16 batch rows. h tile lives in wave-private LDS (D-layout ->
// A-layout transpose via LDS, no barriers needed: DS is in-order per wave).
// x tiles double-buffered with global_load_async_to_lds_b128 (ASYNCcnt);
// async *loads* complete in order, so "s_wait_asynccnt 1" after issuing the
// next prefetch guarantees the current buffer is resident.
// Time loop unrolled x2 so buffer selection is compile-time and all LDS
// gather addresses are loop-invariant registers.

#define SEQ   2048
#define BATCH 4096
#define INF_  3
#define HID   5

typedef __attribute__((ext_vector_type(2))) float v2f;
typedef __attribute__((ext_vector_type(8))) float v8f;

__device__ __forceinline__ float tanh_approx(float x) {
#if __has_builtin(__builtin_amdgcn_tanhf)
  return __builtin_amdgcn_tanhf(x);                 // HW V_TANH_F32 (1 trans op)
#elif __has_builtin(__builtin_amdgcn_tanh_f32)
  return __builtin_amdgcn_tanh_f32(x);
#else
  // tanh(x) = sign(x) * (1 - 2/(exp(2|x|)+1)); branch-free, overflow-safe.
  float ax = __builtin_fabsf(x);
  float e  = __expf(2.0f * ax);
  float r  = __builtin_amdgcn_rcpf(e + 1.0f);
  return __builtin_copysignf(1.0f - 2.0f * r, x);
#endif
}

// Async copy 16B/lane global -> LDS (CDNA5 VGLOBAL async op, ASYNCcnt).
__device__ __forceinline__ void async_b128_to_lds(const float* gsrc, unsigned lds_byte_off) {
  asm volatile("global_load_async_to_lds_b128 %0, %1, off"
               :: "v"(lds_byte_off),
                  "v"((unsigned long long)(uintptr_t)gsrc)
               : "memory");
}

__device__ __forceinline__ v8f wmma_f32_k4(v2f a, v2f b, v8f c) {
  return __builtin_amdgcn_wmma_f32_16x16x4_f32(false, a, false, b,
                                               (short)0, c, false, false);
}

__global__ void __launch_bounds__(64) rnn_fused_wmma(
    const float* __restrict__ x,    // (S,B,IN)
    const float* __restrict__ h0,   // (1,B,HID)
    const float* __restrict__ Wih,  // (HID,IN)
    const float* __restrict__ Whh,  // (HID,HID)
    const float* __restrict__ bih,  // (HID)
    const float* __restrict__ bhh,  // (HID)
    float* __restrict__ out)        // ys (S,B,HID) then h_n (1,B,HID)
{
  // Per-wave LDS slice: xbuf0[48] | xbuf1[48] | hbuf[80] | pad -> 192 floats.
  __shared__ __align__(16) float smem[2 * 192];

  const int lane  = threadIdx.x & 31;
  const int wave  = threadIdx.x >> 5;
  const int b0    = blockIdx.x * 32 + wave * 16;   // 16 batches per wave
  const int wbase = wave * 192;
  const int xoff0 = wbase, xoff1 = wbase + 48, hoff = wbase + 96;

  const int  n  = lane & 15;       // N (hidden j) for B/C/D layouts
  const int  m  = n;               // M (batch-in-tile) for A layout
  const int  hi = lane >> 4;       // lane half: K pair select / M+8
  const bool nv = (n < HID);
  const int  nc = nv ? n : 0;

  // --- B tiles (zero-padded cols n>=HID) + bias C tile. K order: x0..x2,h0..h4.
  const float wi0 = Wih[nc*3+0], wi1 = Wih[nc*3+1], wi2 = Wih[nc*3+2];
  const float wh0 = Whh[nc*5+0], wh1 = Whh[nc*5+1], wh2 = Whh[nc*5+2];
  const float wh3 = Whh[nc*5+3], wh4 = Whh[nc*5+4];
  v2f b1, b2;
  b1.x = nv ? (hi ? wi2 : wi0) : 0.0f;   // K0 = x0 , K2 = x2
  b1.y = nv ? (hi ? wh0 : wi1) : 0.0f;   // K1 = x1 , K3 = h0
  b2.x = nv ? (hi ? wh3 : wh1) : 0.0f;   // K4 = h1 , K6 = h3
  b2.y = nv ? (hi ? wh4 : wh2) : 0.0f;   // K5 = h2 , K7 = h4
  const float bias = nv ? (bih[nc] + bhh[nc]) : 0.0f;
  v8f cb;
#pragma unroll
  for (int r = 0; r < 8; ++r) cb[r] = bias;

  // --- Loop-invariant per-lane addresses. ---
  const int hb    = hoff + m * HID;
  const int a1x0  = xoff0 + m * 3 + 2 * hi;          // x0|x2, buffer 0
  const int a1x1  = xoff1 + m * 3 + 2 * hi;          // x0|x2, buffer 1
  const int a1y0  = hi ? hb : (xoff0 + m * 3 + 1);   // x1|h0, buffer 0
  const int a1y1  = hi ? hb : (xoff1 + m * 3 + 1);   // x1|h0, buffer 1
  const int a2xA  = hb + 1 + 2 * hi;                 // h1|h3
  const int a2yA  = hb + 2 + 2 * hi;                 // h2|h4
  const int hstb  = hoff + 8 * hi * HID + n;         // h store base (D row M=8*hi+r)
  const unsigned ldsx0 = (unsigned)(uintptr_t)(smem + xoff0 + lane * 4);
  const unsigned ldsx1 = (unsigned)(uintptr_t)(smem + xoff1 + lane * 4);
  const float*   xsrc  = x + (size_t)b0 * INF_ + lane * 4;
  const float*   hlds  = smem + hoff + lane * 4;
  const bool pf = (lane < 12);   // 12 lanes x b128 = 48 floats = one x tile
  const bool ov = (lane < 20);   // 20 lanes x b128 = 80 floats = one h tile
  float* outp = out + (size_t)b0 * HID + lane * 4;

  // --- Init h tile from h0; prefetch x tile for s=0 into buffer 0. ---
  if (ov) *(float4*)(smem + hoff + lane * 4) =
            *(const float4*)(h0 + (size_t)b0 * HID + lane * 4);
  if (pf) async_b128_to_lds(xsrc, ldsx0);

#define RNN_STEP(AX, AY, NXT_IDX, LDS_NXT)                                     \
  {                                                                            \
    if (pf) async_b128_to_lds(xsrc + (size_t)(NXT_IDX) * (BATCH * INF_),       \
                              (LDS_NXT));                                      \
    asm volatile("s_wait_asynccnt 0x1" ::: "memory");                          \
    v2f a1, a2;                                                                \
    a1.x = smem[(AX)];  a1.y = smem[(AY)];                                     \
    a2.x = smem[a2xA];  a2.y = smem[a2yA];                                     \
    v8f d = wmma_f32_k4(a1, b1, cb);                                           \
    d     = wmma_f32_k4(a2, b2, d);                                            \
    _Pragma("unroll")                                                          \
    for (int r = 0; r < 8; ++r) {                                              \
      const float hv = tanh_approx(d[r]);                                      \
      if (nv) smem[hstb + r * HID] = hv;                                       \
    }                                                                          \
    if (ov) *(float4*)outp = *(const float4*)hlds;                             \
    outp += (size_t)BATCH * HID;                                               \
  }

  for (int s = 0; s < SEQ; s += 2) {
    // even step: consume buffer 0, prefetch s+1 into buffer 1
    RNN_STEP(a1x0, a1y0, s + 1, ldsx1);
    // odd step: consume buffer 1, prefetch s+2 into buffer 0 (wrap harmless)
    const int nxt2 = (s + 2 < SEQ) ? (s + 2) : 0;
    RNN_STEP(a1x1, a1y1, nxt2, ldsx0);
  }
#undef RNN_STEP

  // --- h_n output ---
  if (ov) *(float4*)(out + (size_t)SEQ * BATCH * HID + (size_t)b0 * HID + lane * 4) =
            *(const float4*)hlds;
}

extern "C" void kernel_launch(void* const* d_in, const int* in_sizes, int n_in,
                              void* d_out, int out_size, void* d_ws, size_t ws_size,
                              hipStream_t stream) {
  (void)in_sizes; (void)n_in; (void)out_size; (void)d_ws; (void)ws_size;
  const float* x   = (const float*)d_in[0];
  const float* h0  = (const float*)d_in[1];
  const float* Wih = (const float*)d_in[2];
  const float* Whh = (const float*)d_in[3];
  const float* bih = (const float*)d_in[4];
  const float* bhh = (const float*)d_in[5];
  float* out = (float*)d_out;
  dim3 grid(BATCH / 32);   // 2 waves/block * 16 batches/wave = 256 waves total
  dim3 block(64);
  hipLaunchKernelGGL(rnn_fused_wmma, grid, block, 0, stream,
                     x, h0, Wih, Whh, bih, bhh, out);
}